// HardMoEClassifier_24842090840420
// MI455X (gfx1250) — compile-verified
//
#include <hip/hip_runtime.h>

// HardMoE classifier head: fused skinny GEMM [B,768] x [768,32] (two 16-wide
// WMMA N-tiles: cols 0..5 gate logits, cols 16..27 all-expert outputs) +
// per-row argmax routing. HBM-bound: 403 MB of cls_token read exactly once,
// staged through LDS by the Tensor Data Mover. Four independent WMMA
// accumulation chains hide f32-WMMA latency at low occupancy.

typedef __attribute__((ext_vector_type(2))) float v2f;
typedef __attribute__((ext_vector_type(8))) float v8f;
typedef __attribute__((ext_vector_type(4))) unsigned int v4u;
typedef __attribute__((ext_vector_type(4))) int v4i;
typedef __attribute__((ext_vector_type(8))) int v8i;

#define B_ROWS 131072
#define D_DIM  768
#define E_EXP  6
#define L_LAB  2
#define N_COLS 32            // padded N: [0,6)=gate, [16,28)=experts, rest zero
#define ROWS_PER_BLOCK 128
#define THREADS 256
#define WAVES   8

#define KC      64           // K-chunk staged per TDM tile (per wave: 16 x 64 f32)
#define NCHUNK  (D_DIM / KC) // 12
#define ARS     66           // padded A-row stride in dwords (64 + 2 pad)
#define ABUF_FLOATS (16 * ARS)            // 1056 dwords per A buffer
#define W_FLOATS    (D_DIM * N_COLS)      // 24576 dwords of packed weights
#define LDS_FLOATS  (W_FLOATS + WAVES * 2 * ABUF_FLOATS)   // 41472
#define LDS_BYTES   (LDS_FLOATS * 4)                       // 165888

// TDM D# group1 word0 flag sets (data_size=4B at [17:16]; pad cfg for A tiles:
// pad_enable bit20, pad_interval=5 (64 dwords) bits[24:22], pad_amount=1 (2 dwords) bits[31:25]).
#define TDM_FLAGS_PLAIN  (2u << 16)
#define TDM_FLAGS_APAD   ((2u << 16) | (1u << 20) | (5u << 22) | (1u << 25))

#if __has_include(<hip/amd_detail/amd_gfx1250_TDM.h>)
#define TDM_SIX_ARG 1
#else
#define TDM_SIX_ARG 0
#endif

// Issue one TENSOR_LOAD_TO_LDS for a 2D tile (tile1 rows x tile0 dwords,
// row stride stride_dw dwords) from gptr into LDS byte offset lds_bytes.
__device__ __forceinline__ void tdm_load(const void* gptr, unsigned lds_bytes,
                                         unsigned td0, unsigned td1,
                                         unsigned tile0, unsigned tile1,
                                         unsigned stride_dw, unsigned flags) {
  unsigned long long ga = (unsigned long long)(size_t)gptr;
  v4u g0 = { 1u,                                  // count=1 (valid user D#)
             lds_bytes,                           // lds_addr
             (unsigned)ga,                        // global_addr[31:0]
             ((unsigned)(ga >> 32) & 0x01FFFFFFu) | (2u << 30) };  // [56:32] | type=2
  v8i g1 = { (int)flags,
             (int)((td0 & 0xFFFFu) << 16),                         // tensor_dim0 lo16
             (int)(((td0 >> 16) & 0xFFFFu) | ((td1 & 0xFFFFu) << 16)),
             (int)(((td1 >> 16) & 0xFFFFu) | (tile0 << 16)),       // tile_dim0
             (int)(tile1 & 0xFFFFu),                               // tile_dim1 (tile_dim2=0)
             (int)stride_dw,                                       // tensor_dim0_stride lo32
             0, 0 };
  v4i z4 = {0, 0, 0, 0};
#if TDM_SIX_ARG
  v8i z8 = {0, 0, 0, 0, 0, 0, 0, 0};
  __builtin_amdgcn_tensor_load_to_lds(g0, g1, z4, z4, z8, 0);
#else
  __builtin_amdgcn_tensor_load_to_lds(g0, g1, z4, z4, 0);
#endif
}

// Pack weights into WMMA-B-fragment swizzle: (k, j) -> Wp[((k>>1)*32 + j)*2 + (k&1)]
__global__ void pack_weights_kernel(const float* __restrict__ gate_w,
                                    const float* __restrict__ expert_w,
                                    float* __restrict__ Wp) {
  int k = blockIdx.x * blockDim.x + threadIdx.x;
  if (k >= D_DIM) return;
  #pragma unroll
  for (int j = 0; j < N_COLS; ++j) {
    float v = 0.0f;
    if (j < E_EXP) {
      v = gate_w[j * D_DIM + k];                                  // gate_w [E,D]
    } else if (j >= 16 && j < 16 + E_EXP * L_LAB) {
      int jj = j - 16;                                            // expert_w [E,D,L]
      v = expert_w[(jj >> 1) * (D_DIM * L_LAB) + k * L_LAB + (jj & 1)];
    }
    Wp[((k >> 1) * N_COLS + j) * 2 + (k & 1)] = v;
  }
}

__global__ __launch_bounds__(THREADS, 1)
void moe_head_kernel(const float* __restrict__ cls,
                     const float* __restrict__ gate_b,
                     const float* __restrict__ expert_b,
                     const float* __restrict__ Wp,
                     float* __restrict__ out) {
  extern __shared__ float smem[];       // [0,24576) weights; then per-wave A buffers
  const int tid  = threadIdx.x;
  const int lane = tid & 31;
  const int wave = tid >> 5;

  const int rowbase = blockIdx.x * ROWS_PER_BLOCK + wave * 16;
  const int r     = lane & 15;          // A/C row within tile
  const int khalf = lane >> 4;          // which K-pair this lane holds
  const int col   = lane & 15;          // B/C column within a 16-wide N tile

  const unsigned a0 = W_FLOATS + (unsigned)wave * 2u * ABUF_FLOATS;
  const unsigned a1 = a0 + ABUF_FLOATS;
  const float* gtile = cls + (size_t)rowbase * D_DIM;

  // Wave 0: TDM the whole packed-weight block (96 KB, contiguous 1D) into LDS.
  if (wave == 0)
    tdm_load(Wp, 0u, W_FLOATS, 1u, W_FLOATS, 1u, W_FLOATS, TDM_FLAGS_PLAIN);
  // All waves: TDM chunk 0 of their private 16-row A tile (16 x 64 f32, padded rows).
  tdm_load(gtile, a0 * 4u, D_DIM, B_ROWS, KC, 16u, D_DIM, TDM_FLAGS_APAD);
  if (wave == 0)
    __builtin_amdgcn_s_wait_tensorcnt(1);   // weight DMA (oldest) complete
  __syncthreads();                          // publish weights to all waves

  // Four independent accumulation chains (gate/expert x even/odd K-step).
  v8f acc0e = {}, acc0o = {};               // gate tile (cols 0..15)
  v8f acc1e = {}, acc1o = {};               // expert tile (cols 16..31)

  for (int c = 0; c < NCHUNK; ++c) {
    if (c + 1 < NCHUNK) {                   // prefetch next chunk into other buffer
      tdm_load(gtile + (c + 1) * KC, ((c & 1) ? a0 : a1) * 4u,
               D_DIM, B_ROWS, KC, 16u, D_DIM, TDM_FLAGS_APAD);
      __builtin_amdgcn_s_wait_tensorcnt(1); // current chunk's DMA complete
    } else {
      __builtin_amdgcn_s_wait_tensorcnt(0);
    }

    const unsigned ab = (c & 1) ? a1 : a0;
    const float* arow = &smem[ab + (unsigned)r * ARS + 2u * (unsigned)khalf];
    #pragma unroll
    for (int kk = 0; kk < KC; kk += 8) {
      // Two K=4 steps per iteration, feeding the even/odd chains.
      v2f ae = *(const v2f*)(arow + kk);
      v2f ao = *(const v2f*)(arow + kk + 4);
      const int p = ((c * KC + kk) >> 1) + khalf;    // K-pair index (even step)
      const v2f* we = (const v2f*)&smem[p * N_COLS * 2];
      const v2f* wo = (const v2f*)&smem[(p + 2) * N_COLS * 2];
      v2f b0e = we[col];
      v2f b1e = we[col + 16];
      v2f b0o = wo[col];
      v2f b1o = wo[col + 16];
      acc0e = __builtin_amdgcn_wmma_f32_16x16x4_f32(false, ae, false, b0e,
                                                    (short)0, acc0e, false, false);
      acc1e = __builtin_amdgcn_wmma_f32_16x16x4_f32(false, ae, false, b1e,
                                                    (short)0, acc1e, false, false);
      acc0o = __builtin_amdgcn_wmma_f32_16x16x4_f32(false, ao, false, b0o,
                                                    (short)0, acc0o, false, false);
      acc1o = __builtin_amdgcn_wmma_f32_16x16x4_f32(false, ao, false, b1o,
                                                    (short)0, acc1o, false, false);
    }
    // A buffer fully consumed before TDM may overwrite it next iteration.
    asm volatile("s_wait_dscnt 0x0" ::: "memory");
  }

  // Merge even/odd chains.
  v8f acc0, acc1;
  #pragma unroll
  for (int i = 0; i < 8; ++i) {
    acc0[i] = acc0e[i] + acc0o[i];
    acc1[i] = acc1e[i] + acc1o[i];
  }

  __syncthreads();                          // all waves done with weights -> reuse LDS

  // Spill C fragments: VGPR i holds (M = i + 8*(lane/16), N = lane%16).
  float* epil = smem + wave * (16 * N_COLS);
  const int mrow = 8 * khalf;
  #pragma unroll
  for (int i = 0; i < 8; ++i) {
    epil[(i + mrow) * N_COLS + col]      = acc0[i];
    epil[(i + mrow) * N_COLS + col + 16] = acc1[i];
  }
  __syncthreads();

  // Lanes 0..15: per-row argmax over 6 gates (first-max tiebreak) + routed select.
  if (lane < 16) {
    const float* rowv = epil + r * N_COLS;
    int   best = 0;
    float bv   = rowv[0] + gate_b[0];
    #pragma unroll
    for (int j = 1; j < E_EXP; ++j) {
      float v = rowv[j] + gate_b[j];
      if (v > bv) { bv = v; best = j; }
    }
    const size_t orow = (size_t)(rowbase + r) * L_LAB;
    out[orow + 0] = rowv[16 + best * 2 + 0] + expert_b[best * 2 + 0];
    out[orow + 1] = rowv[16 + best * 2 + 1] + expert_b[best * 2 + 1];
  }
}

extern "C" void kernel_launch(void* const* d_in, const int* in_sizes, int n_in,
                              void* d_out, int out_size, void* d_ws, size_t ws_size,
                              hipStream_t stream) {
  const float* cls      = (const float*)d_in[0];   // [B, D]
  const float* gate_w   = (const float*)d_in[1];   // [E, D]
  const float* gate_b   = (const float*)d_in[2];   // [E]
  const float* expert_w = (const float*)d_in[3];   // [E, D, L]
  const float* expert_b = (const float*)d_in[4];   // [E, L]
  float* out = (float*)d_out;                      // [B, L]
  float* Wp  = (float*)d_ws;                       // 768*32 floats swizzled weights

  pack_weights_kernel<<<(D_DIM + 255) / 256, 256, 0, stream>>>(gate_w, expert_w, Wp);

  dim3 grid(B_ROWS / ROWS_PER_BLOCK);
  moe_head_kernel<<<grid, THREADS, LDS_BYTES, stream>>>(cls, gate_b, expert_b, Wp, out);
}